// RankLoss_25099788878503
// MI455X (gfx1250) — compile-verified
//
#include <hip/hip_runtime.h>
#include <hip/hip_bf16.h>
#include <math.h>

typedef __attribute__((ext_vector_type(2))) float v2f;
typedef __attribute__((ext_vector_type(8))) float v8f;

#define N_TOTAL     8192
#define BLOCK       256
#define WAVES       (BLOCK / 32)
#define I_PER_BLOCK (WAVES * 16)          // 128
#define J_CHUNK     512
#define IB          (N_TOTAL / I_PER_BLOCK)  // 64
#define JBLK        (N_TOTAL / J_CHUNK)      // 16
#define NBLOCKS     (IB * JBLK)              // 1024

// diff[i][j] = p_i - p_j via two V_WMMA_F32_16X16X4_F32:
//   dP = A(col0 = p_i) x B(row0 = ones) + 0      -> dP[m][n] = p_i[m]   (hoisted)
//   d  = A(col0 = ones) x B(row0 = -p_j) + dP    -> d[m][n]  = p_i[m] - p_j[n]
// Bit-exact vs v_sub_f32 (products exact, single rounding on the accumulate).
__global__ __launch_bounds__(BLOCK)
void rankloss_main(const float* __restrict__ preds,
                   const float* __restrict__ tgts,
                   double* __restrict__ bsum,
                   unsigned int* __restrict__ bcnt) {
    __shared__ float s_pj[J_CHUNK];
    __shared__ float s_tj[J_CHUNK];
    __shared__ float s_wsum[WAVES];
    __shared__ unsigned int s_wcnt[WAVES];

    const int tid  = threadIdx.x;
    const int lane = tid & 31;
    const int wave = tid >> 5;
    const int i0   = blockIdx.x * I_PER_BLOCK + wave * 16;
    const int j0   = blockIdx.y * J_CHUNK;

    // Stage this block's j-chunk (predictions and targets) into LDS.
    for (int k = tid; k < J_CHUNK; k += BLOCK) {
        s_pj[k] = preds[j0 + k];
        s_tj[k] = tgts[j0 + k];
    }

    // Per-lane row targets: lane covers rows m = mbase + r (r = 0..7), per C/D layout.
    const int mbase = 8 * (lane >> 4);
    float ti[8];
#pragma unroll
    for (int r = 0; r < 8; ++r) ti[r] = tgts[i0 + mbase + r];

    const float pi_lane = preds[i0 + (lane & 15)];  // per-lane p_i[lane % 16], read in all lanes
    const bool  lo16    = (lane < 16);

    v2f a_p;   a_p.x   = lo16 ? pi_lane : 0.0f; a_p.y   = 0.0f;
    v2f a_one; a_one.x = lo16 ? 1.0f    : 0.0f; a_one.y = 0.0f;
    v2f b_one = a_one;
    v8f zero  = {0.f, 0.f, 0.f, 0.f, 0.f, 0.f, 0.f, 0.f};

    // dP[m][n] = p_i[m] (loop-invariant; one WMMA per wave)
    v8f dP = __builtin_amdgcn_wmma_f32_16x16x4_f32(
        false, a_p, false, b_one, (short)0, zero, false, false);

    __syncthreads();

    const float LOG2E = 1.4426950408889634f;
    const float LN2   = 0.6931471805599453f;

    float        lsum = 0.0f;
    unsigned int lcnt = 0u;
    const int    ncol = lane & 15;

    for (int jb = 0; jb < J_CHUNK; jb += 16) {
        // Unpredicated LDS reads in ALL lanes (valid addresses), then cndmask:
        // keeps EXEC all-ones around the WMMA (ISA requirement) and avoids
        // exec save/restore per iteration.
        const float pj  = s_pj[jb + ncol];
        const float tjv = s_tj[jb + ncol];
        v2f b2;
        b2.x = lo16 ? -pj : 0.0f;
        b2.y = 0.0f;

        // d[m][n] = p_i[m] - p_j[n]
        v8f d = __builtin_amdgcn_wmma_f32_16x16x4_f32(
            false, a_one, false, b2, (short)0, dP, false, false);

#pragma unroll
        for (int r = 0; r < 8; ++r) {
            const float x  = d[r];
            // log_sigmoid(x) = min(x,0) - ln2 * log2(1 + exp2(-log2e*|x|))
            const float e  = __builtin_amdgcn_exp2f(-LOG2E * __builtin_fabsf(x));
            const float lg = __builtin_amdgcn_logf(1.0f + e);   // v_log_f32 (log2)
            const float ls = __builtin_fmaf(-LN2, lg, __builtin_fminf(x, 0.0f));
            const bool  m  = tjv > ti[r];   // mask: targets[j] > targets[i]
            lsum += m ? ls : 0.0f;
            lcnt += m ? 1u : 0u;
        }
    }

    // Deterministic wave reduction (wave32 shuffle tree)
#pragma unroll
    for (int off = 16; off > 0; off >>= 1) {
        lsum += __shfl_down(lsum, off, 32);
        lcnt += (unsigned int)__shfl_down((int)lcnt, off, 32);
    }
    if (lane == 0) { s_wsum[wave] = lsum; s_wcnt[wave] = lcnt; }
    __syncthreads();

    if (tid == 0) {
        double       bs = 0.0;
        unsigned int bc = 0u;
#pragma unroll
        for (int w = 0; w < WAVES; ++w) { bs += (double)s_wsum[w]; bc += s_wcnt[w]; }
        const int bid = blockIdx.y * IB + blockIdx.x;
        bsum[bid] = bs;
        bcnt[bid] = bc;
    }
}

__global__ __launch_bounds__(256)
void rankloss_finalize(const double* __restrict__ bsum,
                       const unsigned int* __restrict__ bcnt,
                       float* __restrict__ out, int nb) {
    __shared__ double    ssum[256];
    __shared__ long long scnt[256];
    const int tid = threadIdx.x;

    double    s = 0.0;
    long long c = 0;
    for (int b = tid; b < nb; b += 256) {
        s += bsum[b];
        c += (long long)bcnt[b];
    }
    ssum[tid] = s;
    scnt[tid] = c;
    __syncthreads();

    for (int off = 128; off > 0; off >>= 1) {
        if (tid < off) {
            ssum[tid] += ssum[tid + off];
            scnt[tid] += scnt[tid + off];
        }
        __syncthreads();
    }
    if (tid == 0) out[0] = (float)(ssum[0] / (double)scnt[0]);
}

extern "C" void kernel_launch(void* const* d_in, const int* in_sizes, int n_in,
                              void* d_out, int out_size, void* d_ws, size_t ws_size,
                              hipStream_t stream) {
    const float* preds = (const float*)d_in[0];
    const float* tgts  = (const float*)d_in[1];
    float* out = (float*)d_out;

    double*       bsum = (double*)d_ws;                   // NBLOCKS doubles
    unsigned int* bcnt = (unsigned int*)(bsum + NBLOCKS); // NBLOCKS uints

    dim3 grid(IB, JBLK);
    rankloss_main<<<grid, BLOCK, 0, stream>>>(preds, tgts, bsum, bcnt);
    rankloss_finalize<<<1, 256, 0, stream>>>(bsum, bcnt, out, NBLOCKS);
}